// TerrainReconstructionModel_22617297781338
// MI455X (gfx1250) — compile-verified
//
#include <hip/hip_runtime.h>
#include <hip/hip_bf16.h>
#include <math.h>

typedef __attribute__((ext_vector_type(2))) float v2f;
typedef __attribute__((ext_vector_type(8))) float v8f;

// ---------------------------------------------------------------------------
// Weight pre-pack: W[K][Cin][Cout] (row-major) -> fragment order
//   Wp[((k*(Cin/4) + cb)*(Cout/16) + tn)*64 + lane*2 + j]
//     = W[k][cb*4 + 2*(lane>>4) + j][tn*16 + (lane&15)]
// so each lane's B fragment (V_WMMA_F32_16X16X4_F32) is ONE contiguous 8B
// load, fully coalesced across the wave (256B/request).
// ---------------------------------------------------------------------------
__global__ void pack_w(const float* __restrict__ W, float* __restrict__ Wp,
                       int K, int Cin, int Cout)
{
    int i = blockIdx.x * blockDim.x + threadIdx.x;          // one pair / thread
    int nT = Cout >> 4, CB = Cin >> 2;
    int total = K * CB * nT * 32;
    if (i >= total) return;
    int lane = i & 31;
    int rest = i >> 5;
    int tn = rest % nT;  rest /= nT;
    int cb = rest % CB;
    int k  = rest / CB;
    int c   = cb * 4 + ((lane >> 4) << 1);
    int col = tn * 16 + (lane & 15);
    const float* src = W + ((size_t)k * Cin + c) * Cout + col;
    Wp[(size_t)i * 2 + 0] = src[0];
    Wp[(size_t)i * 2 + 1] = src[Cout];
}

// ---------------------------------------------------------------------------
// Sparse-conv gather-GEMM on the CDNA5 tensor path:
//   Y[n, o] = sum_k sum_c Xpad[nbr[n,k], c] * W[k, c, o]
// One wave owns 16 points x (TN*16) out-channels: one gathered A fragment
// feeds TN WMMAs (TN accumulators), amortizing the divergent gather.
// A fragment (16x4 f32, ISA layout): lanes 0-15 rows M, K=c+0/c+1 in v.x/v.y;
// lanes 16-31 hold K=c+2/c+3.  B comes pre-packed (see pack_w).
// D (16x16 f32): vgpr i = row i (lanes 0-15) / row 8+i (lanes 16-31).
// Index == Nin selects the implicit zero pad row; branches re-converge before
// the WMMA so EXEC is all-ones at the matrix op as required.
// ---------------------------------------------------------------------------
template <int TN>
__global__ __launch_bounds__(32)
void sconv_wmma_f32(const float* __restrict__ X, const int* __restrict__ nbr,
                    const float* __restrict__ Wp, float* __restrict__ Y,
                    int N, int Nin, int K, int Cin, int Cout)
{
    const int lane  = threadIdx.x;       // 0..31
    const int half  = lane >> 4;         // 0 | 1
    const int l15   = lane & 15;
    const int tileM = blockIdx.x << 4;
    const int tile0 = blockIdx.y * TN;   // first cout tile handled by this wave
    const int nT    = Cout >> 4;
    const int CB    = Cin >> 2;
    const int row   = tileM + l15;
    const bool rowOK = (row < N);

    v8f acc[TN];
#pragma unroll
    for (int t = 0; t < TN; ++t)
        acc[t] = (v8f){0.f, 0.f, 0.f, 0.f, 0.f, 0.f, 0.f, 0.f};

    const int aoff = (half << 1);

    for (int k = 0; k < K; ++k) {
        int r = rowOK ? nbr[(size_t)row * K + k] : Nin;     // OOB rows -> pad
        const bool valid = (r < Nin);                       // r==Nin -> zero row
        const float* xr = X + (size_t)r * Cin;

        // Hide the next tap's gather latency behind this tap's c-loop.
        if (rowOK && (k + 1 < K)) {
            int r2 = nbr[(size_t)row * K + k + 1];
            if (r2 < Nin)
                __builtin_prefetch(X + (size_t)r2 * Cin + aoff, 0, 1);
        }

        const float* wb = Wp + ((size_t)k * CB * nT + tile0) * 64 + lane * 2;

        for (int cb = 0; cb < CB; ++cb) {
            v2f a = {0.f, 0.f};
            if (valid) a = *(const v2f*)(xr + (cb << 2) + aoff);   // 8B gather

#pragma unroll
            for (int t = 0; t < TN; ++t) {
                v2f b = *(const v2f*)(wb + t * 64);                // coalesced 8B
                acc[t] = __builtin_amdgcn_wmma_f32_16x16x4_f32(
                    /*neg_a=*/false, a, /*neg_b=*/false, b,
                    /*c_mod=*/(short)0, acc[t],
                    /*reuse_a=*/false, /*reuse_b=*/false);
            }
            wb += (size_t)nT * 64;
        }
    }

#pragma unroll
    for (int t = 0; t < TN; ++t) {
        int col = (tile0 + t) * 16 + l15;
#pragma unroll
        for (int i = 0; i < 8; ++i) {
            int m = tileM + (half ? (8 + i) : i);
            if (m < N) Y[(size_t)m * Cout + col] = acc[t][i];
        }
    }
}

// ---------------------------------------------------------------------------
// Deterministic BatchNorm: fixed 64-block partial sums (thread == channel,
// coalesced), fixed-order finalize, then fused scale/shift + ReLU apply that
// can write into a strided concat buffer.
// ---------------------------------------------------------------------------
__global__ void bn_partial(const float* __restrict__ raw, int N, int C,
                           float* __restrict__ stats)
{
    int c = threadIdx.x;              // blockDim.x == C
    int blk = blockIdx.x;
    float s = 0.f, sq = 0.f;
    for (int n = blk; n < N; n += gridDim.x) {
        float v = raw[(size_t)n * C + c];
        s += v; sq += v * v;
    }
    stats[(size_t)(2 * blk) * C + c]     = s;
    stats[(size_t)(2 * blk + 1) * C + c] = sq;
}

__global__ void bn_finalize(const float* __restrict__ stats, int NB, int N, int C,
                            const float* __restrict__ g, const float* __restrict__ b,
                            float* __restrict__ ss)
{
    int c = threadIdx.x;              // blockDim.x == C
    float s = 0.f, sq = 0.f;
    for (int i = 0; i < NB; ++i) {    // fixed order -> deterministic
        s  += stats[(size_t)(2 * i) * C + c];
        sq += stats[(size_t)(2 * i + 1) * C + c];
    }
    float mu  = s / (float)N;
    float var = sq / (float)N - mu * mu;
    if (var < 0.f) var = 0.f;
    float sc = g[c] * rsqrtf(var + 1e-5f);
    ss[c]     = sc;
    ss[C + c] = b[c] - mu * sc;
}

__global__ void bn_relu_apply(const float* __restrict__ raw, int N, int C,
                              const float* __restrict__ ss,
                              float* __restrict__ dst, int dstStride, int colOff)
{
    int i = blockIdx.x * blockDim.x + threadIdx.x;
    if (i >= N * C) return;
    int n = i / C, c = i - n * C;
    float v = raw[i] * ss[c] + ss[C + c];
    dst[(size_t)n * dstStride + colOff + c] = v > 0.f ? v : 0.f;
}

// Skip-connection gather (with zero pad row) into the concat buffer.
__global__ void gather_pad_k(const float* __restrict__ src, int NinSrc, int C,
                             const int* __restrict__ idx, int N,
                             float* __restrict__ dst, int dstStride, int colOff)
{
    int i = blockIdx.x * blockDim.x + threadIdx.x;
    if (i >= N * C) return;
    int n = i / C, c = i - n * C;
    int r = idx[n];
    float v = (r < NinSrc) ? src[(size_t)r * C + c] : 0.f;
    dst[(size_t)n * dstStride + colOff + c] = v;
}

// Occupancy head: logits[n] = d[n,:] . w + b   (w is (C,1))
__global__ void occ_head(const float* __restrict__ d, int N, int C,
                         const float* __restrict__ w, const float* __restrict__ b,
                         float* __restrict__ out)
{
    int n = blockIdx.x * blockDim.x + threadIdx.x;
    if (n >= N) return;
    const float* dr = d + (size_t)n * C;
    float s = b[0];
    for (int c = 0; c < C; ++c) s += dr[c] * w[c];
    out[n] = s;
}

// Deterministic any(keep) + first-argmax reduction over logits (single block).
__global__ void prune_reduce(const float* __restrict__ logits, int N,
                             int* __restrict__ flags)
{
    __shared__ float smax[1024];
    __shared__ int   sidx[1024];
    __shared__ int   sany[1024];
    int t = threadIdx.x;
    float bm = -3.402823466e+38f;
    int bi = 0, any = 0;
    for (int i = t; i < N; i += 1024) {
        float l = logits[i];
        if (l >= 0.f) any = 1;                 // sigmoid(l) >= 0.5  <=>  l >= 0
        if (l > bm) { bm = l; bi = i; }        // strict > keeps earliest index
    }
    smax[t] = bm; sidx[t] = bi; sany[t] = any;
    __syncthreads();
    for (int s = 512; s > 0; s >>= 1) {
        if (t < s) {
            if (smax[t + s] > smax[t] ||
                (smax[t + s] == smax[t] && sidx[t + s] < sidx[t])) {
                smax[t] = smax[t + s]; sidx[t] = sidx[t + s];
            }
            sany[t] |= sany[t + s];
        }
        __syncthreads();
    }
    if (t == 0) { flags[0] = sany[0]; flags[1] = sidx[0]; }
}

__global__ void prune_apply(float* __restrict__ d, int N, int C,
                            const float* __restrict__ logits,
                            const int* __restrict__ flags)
{
    int i = blockIdx.x * blockDim.x + threadIdx.x;
    if (i >= N * C) return;
    int n = i / C;
    int keep = flags[0] ? (logits[n] >= 0.f) : (n == flags[1]);
    if (!keep) d[i] = 0.f;
}

// Offset head: sigmoid(d @ W(32x3) + b)
__global__ void off_head(const float* __restrict__ d, int N, int C,
                         const float* __restrict__ w, const float* __restrict__ b,
                         float* __restrict__ out)
{
    int n = blockIdx.x * blockDim.x + threadIdx.x;
    if (n >= N) return;
    const float* dr = d + (size_t)n * C;
    for (int j = 0; j < 3; ++j) {
        float s = b[j];
        for (int c = 0; c < C; ++c) s += dr[c] * w[c * 3 + j];
        out[(size_t)n * 3 + j] = 1.f / (1.f + expf(-s));
    }
}

// ---------------------------------------------------------------------------
// Host orchestration
// ---------------------------------------------------------------------------
extern "C" void kernel_launch(void* const* d_in, const int* in_sizes, int n_in,
                              void* d_out, int out_size, void* d_ws, size_t ws_size,
                              hipStream_t stream)
{
    (void)in_sizes; (void)n_in; (void)out_size; (void)ws_size;

    static const int NS[5]  = {40000, 20000, 10000, 5000, 2500};
    static const int ENC[4] = {64, 128, 192, 256};
    static const int DEC[4] = {128, 96, 64, 32};

    const float* x_feat = (const float*)d_in[0];
    const int* idx_stem = (const int*)d_in[1];
    const int* idx_down[4] = {(const int*)d_in[2], (const int*)d_in[4],
                              (const int*)d_in[6], (const int*)d_in[8]};
    const int* idx_ref[4]  = {(const int*)d_in[3], (const int*)d_in[5],
                              (const int*)d_in[7], (const int*)d_in[9]};
    const int *idx_up[4], *idx_skip[4], *idx_fuse[4];
    for (int t = 0; t < 4; ++t) {           // inputs 10..21: up/skip/fuse for lvl 3..0
        int lvl = 3 - t;
        idx_up[lvl]   = (const int*)d_in[10 + 3 * t + 0];
        idx_skip[lvl] = (const int*)d_in[10 + 3 * t + 1];
        idx_fuse[lvl] = (const int*)d_in[10 + 3 * t + 2];
    }
    const float* P[83];
    for (int i = 22; i < 83; ++i) P[i] = (const float*)d_in[i];

    // ---- workspace arena (floats) ----
    float* wsf = (float*)d_ws;
    float* F[5];                              // encoder features (post BN+ReLU)
    F[0] = wsf + 0;                           // 40000 x 32
    F[1] = wsf + 1280000;                     // 20000 x 64
    F[2] = wsf + 2560000;                     // 10000 x 128
    F[3] = wsf + 3840000;                     //  5000 x 192
    F[4] = wsf + 4800000;                     //  2500 x 256
    float* RAW   = wsf + 5440000;             // pre-BN conv output (<=1.28M)
    float* CAT   = wsf + 6750720;             // concat / temp activations (<=2.56M)
    float* DA    = wsf + 9372160;             // decoder ping
    float* DB    = wsf + 10682880;            // decoder pong
    float* STATS = wsf + 11993600;            // 64 * 2 * 256
    float* SS    = wsf + 12026368;            // scale/shift (2*256)
    int*   FLAGS = (int*)(wsf + 12026880);    // [anyKeep, argmaxIdx]
    float* WPACK = wsf + 12027008;            // fragment-packed weights (<=5.31M)

    auto sconv = [&](const float* X, const int* nbr, const float* W, float* Y,
                     int N, int Nin, int K, int Cin, int Cout) {
        // Pack this conv's weights into fragment order (stream-ordered).
        int pairs = K * (Cin >> 2) * (Cout >> 4) * 32;
        pack_w<<<(pairs + 255) / 256, 256, 0, stream>>>(W, WPACK, K, Cin, Cout);

        int tiles = Cout >> 4;
        dim3 g((N + 15) / 16, 1);
        if ((tiles & 3) == 0) {
            g.y = tiles >> 2;
            sconv_wmma_f32<4><<<g, 32, 0, stream>>>(X, nbr, WPACK, Y, N, Nin, K, Cin, Cout);
        } else if ((tiles & 1) == 0) {
            g.y = tiles >> 1;
            sconv_wmma_f32<2><<<g, 32, 0, stream>>>(X, nbr, WPACK, Y, N, Nin, K, Cin, Cout);
        } else {
            g.y = tiles;
            sconv_wmma_f32<1><<<g, 32, 0, stream>>>(X, nbr, WPACK, Y, N, Nin, K, Cin, Cout);
        }
    };
    auto bn = [&](const float* raw, int N, int C, const float* g_, const float* b_,
                  float* dst, int stride, int coloff) {
        bn_partial<<<64, C, 0, stream>>>(raw, N, C, STATS);
        bn_finalize<<<1, C, 0, stream>>>(STATS, 64, N, C, g_, b_, SS);
        int tot = N * C;
        bn_relu_apply<<<(tot + 255) / 256, 256, 0, stream>>>(raw, N, C, SS,
                                                             dst, stride, coloff);
    };

    // ---- stem ----
    sconv(x_feat, idx_stem, P[22], RAW, NS[0], NS[0], 81, 4, 32);
    bn(RAW, NS[0], 32, P[23], P[24], F[0], 32, 0);

    // ---- encoder ----
    {
        const int wd[4] = {25, 31, 37, 43};
        const int wr[4] = {28, 34, 40, 46};
        const float* ein = F[0];
        int Cp = 32;
        for (int i = 0; i < 4; ++i) {
            int N = NS[i + 1], Nin = NS[i], C = ENC[i];
            sconv(ein, idx_down[i], P[wd[i]], RAW, N, Nin, 16, Cp, C);
            bn(RAW, N, C, P[wd[i] + 1], P[wd[i] + 2], CAT, C, 0);
            sconv(CAT, idx_ref[i], P[wr[i]], RAW, N, N, 81, C, C);
            bn(RAW, N, C, P[wr[i] + 1], P[wr[i] + 2], F[i + 1], C, 0);
            ein = F[i + 1]; Cp = C;
        }
    }

    // ---- decoder ----
    float* outf = (float*)d_out;
    const int wu[4]  = {49, 57, 65, 73};
    const int wf[4]  = {52, 60, 68, 76};
    const int ocw[4] = {55, 63, 71, 79};
    const int occ_off[4] = {120000, 125000, 135000, 155000}; // lvl 3,2,1,0
    const int Cf[4] = {32, 64, 128, 192};                    // skip channels by lvl

    const float* d = F[4];
    int Cd = 256;
    float* dbuf[2] = {DA, DB};
    for (int t = 0; t < 4; ++t) {
        int lvl = 3 - t;
        int N = NS[lvl], Nin = NS[lvl + 1];
        int cout = DEC[t], cskip = Cf[lvl], ctot = cout + cskip;
        float* dn = dbuf[t & 1];

        sconv(d, idx_up[lvl], P[wu[t]], RAW, N, Nin, 16, Cd, cout);
        bn(RAW, N, cout, P[wu[t] + 1], P[wu[t] + 2], CAT, ctot, 0);
        gather_pad_k<<<(N * cskip + 255) / 256, 256, 0, stream>>>(
            F[lvl], N, cskip, idx_skip[lvl], N, CAT, ctot, cout);
        sconv(CAT, idx_fuse[lvl], P[wf[t]], RAW, N, N, 81, ctot, cout);
        bn(RAW, N, cout, P[wf[t] + 1], P[wf[t] + 2], dn, cout, 0);

        float* logits = outf + occ_off[t];
        occ_head<<<(N + 255) / 256, 256, 0, stream>>>(dn, N, cout,
                                                      P[ocw[t]], P[ocw[t] + 1], logits);
        prune_reduce<<<1, 1024, 0, stream>>>(logits, N, FLAGS);
        prune_apply<<<(N * cout + 255) / 256, 256, 0, stream>>>(dn, N, cout,
                                                                logits, FLAGS);
        d = dn; Cd = cout;
    }

    // ---- offsets head ----
    off_head<<<(NS[0] + 255) / 256, 256, 0, stream>>>(d, NS[0], 32, P[81], P[82], outf);
}